// flow_fusion_4398046511721
// MI455X (gfx1250) — compile-verified
//
#include <hip/hip_runtime.h>

typedef __attribute__((ext_vector_type(2))) float v2f;
typedef __attribute__((ext_vector_type(8))) float v8f;

#define B_    4
#define N_    8192
#define M_    8192
#define C_    128
#define KNN   16
#define ROW_PAD 20                 // 16 data floats + 4 pad; 80B keeps 16B alignment
#define WAVE_LDS (32 * ROW_PAD)

__global__ __launch_bounds__(128) void knn_interp_wmma(
    const float* __restrict__ xyz,      // (B, N, 3)
    const float* __restrict__ new_xyz,  // (B, M, 3)
    const float* __restrict__ feat,     // (B, C, M)
    float* __restrict__ out)            // (B, C, N)
{
    __shared__ __align__(16) float lds[4 * WAVE_LDS];

    const int lane = threadIdx.x & 31;
    const int warp = threadIdx.x >> 5;
    const int wid  = blockIdx.x * 4 + warp;      // global wave id, owns 32 query rows
    const int tilesPerB = N_ / 32;
    const int b  = wid / tilesPerB;
    const int n0 = (wid % tilesPerB) * 32;

    float* wl = lds + warp * WAVE_LDS;

    const float* xb = xyz     + (size_t)b * N_ * 3;
    const float* yb = new_xyz + (size_t)b * M_ * 3;

    // ---- stage ||a||^2 for the 32 owned rows through LDS ----
    {
        const int r = n0 + lane;
        const float x = xb[r*3+0], y = xb[r*3+1], z = xb[r*3+2];
        wl[lane] = x*x + y*y + z*z;
    }
    __builtin_amdgcn_wave_barrier();   // DS ops are in-order within a wave

    // C accumulators carry ||a||^2 per output row (16x16 f32 C layout:
    // vgpr r, lanes 0-15 -> row r; lanes 16-31 -> row r+8)
    v8f c0, c1;
    {
        const int half = (lane >= 16) ? 8 : 0;
#pragma unroll
        for (int r = 0; r < 8; ++r) {
            c0[r] = wl[r + half];
            c1[r] = wl[16 + r + half];
        }
    }
    __builtin_amdgcn_wave_barrier();

    // A matrices (16x4 f32): lanes 0-15 hold K=0,1 ; lanes 16-31 hold K=2,3
    // row vector = (x, y, z, 1)
    v2f a0, a1;
    if (lane < 16) {
        const int r0 = n0 + lane, r1 = n0 + 16 + lane;
        a0.x = xb[r0*3+0]; a0.y = xb[r0*3+1];
        a1.x = xb[r1*3+0]; a1.y = xb[r1*3+1];
    } else {
        const int r0 = n0 + lane - 16, r1 = n0 + lane;
        a0.x = xb[r0*3+2]; a0.y = 1.0f;
        a1.x = xb[r1*3+2]; a1.y = 1.0f;
    }

    // register-resident sorted top-16 (ascending), constant indices only
    float hd[KNN]; int hi[KNN];
#pragma unroll
    for (int k = 0; k < KNN; ++k) { hd[k] = 3.0e38f; hi[k] = 0; }

    const int col     = lane & 15;
    const int halfsel = (lane >= 16) ? 1 : 0;
    const int rbase   = halfsel ? 8 : 0;

    for (int m0 = 0; m0 < M_; m0 += 16) {
        // B matrix (4x16 f32): column vector = (-2x', -2y', -2z', ||b||^2)
        const int m = m0 + col;
        const float nx = yb[m*3+0], ny = yb[m*3+1], nz = yb[m*3+2];
        const float sq2 = nx*nx + ny*ny + nz*nz;
        v2f bv;
        bv.x = halfsel ? (-2.0f*nz) : (-2.0f*nx);
        bv.y = halfsel ? sq2        : (-2.0f*ny);

        // D = A*B + C  ->  full squared distances for a 16x16 tile
        v8f d0 = __builtin_amdgcn_wmma_f32_16x16x4_f32(
                     false, a0, false, bv, (short)0, c0, false, false);
        v8f d1 = __builtin_amdgcn_wmma_f32_16x16x4_f32(
                     false, a1, false, bv, (short)0, c1, false, false);

        // transpose 32x16 tile through LDS so each lane owns one query row
#pragma unroll
        for (int r = 0; r < 8; ++r) {
            wl[(rbase + r)      * ROW_PAD + col] = d0[r];
            wl[(16 + rbase + r) * ROW_PAD + col] = d1[r];
        }
        __builtin_amdgcn_wave_barrier();

        const float4* rp = (const float4*)(wl + lane * ROW_PAD);
        const float4 q0 = rp[0], q1 = rp[1], q2 = rp[2], q3 = rp[3];
        const float v[16] = { q0.x,q0.y,q0.z,q0.w, q1.x,q1.y,q1.z,q1.w,
                              q2.x,q2.y,q2.z,q2.w, q3.x,q3.y,q3.z,q3.w };

#pragma unroll
        for (int j = 0; j < 16; ++j) {
            const float cd = v[j];
            if (cd < hd[15]) {                // rare after warm-up; execz-skipped
                const int ci = m0 + j;
#pragma unroll
                for (int p = 15; p >= 1; --p) {
                    const bool repl = hd[p]   > cd;   // original values (top-down)
                    const bool shf  = hd[p-1] > cd;
                    const float nv = shf ? hd[p-1] : cd;
                    const int   ni = shf ? hi[p-1] : ci;
                    hd[p] = repl ? nv : hd[p];
                    hi[p] = repl ? ni : hi[p];
                }
                if (hd[0] > cd) { hd[0] = cd; hi[0] = ci; }
            }
        }
        __builtin_amdgcn_wave_barrier();
    }

    // ---- inverse-distance weights ----
    float w[KNN]; float wsum = 0.0f;
#pragma unroll
    for (int k = 0; k < KNN; ++k) {
        const float d = hd[k] < 1e-10f ? 1e-10f : hd[k];
        w[k] = 1.0f / d;
        wsum += w[k];
    }
    const float inv = 1.0f / wsum;
#pragma unroll
    for (int k = 0; k < KNN; ++k) w[k] *= inv;

    // ---- weighted gather over 128 channels; L2-resident features ----
    const float* fb = feat + (size_t)b * C_ * M_;
    float*       ob = out  + (size_t)b * C_ * N_;
    const int n = n0 + lane;
#pragma unroll 4
    for (int c = 0; c < C_; ++c) {
        const float* fr = fb + (size_t)c * M_;
        float acc = 0.0f;
#pragma unroll
        for (int k = 0; k < KNN; ++k) acc += w[k] * fr[hi[k]];
        ob[(size_t)c * N_ + n] = acc;   // coalesced along n
    }
}

extern "C" void kernel_launch(void* const* d_in, const int* in_sizes, int n_in,
                              void* d_out, int out_size, void* d_ws, size_t ws_size,
                              hipStream_t stream) {
    (void)in_sizes; (void)n_in; (void)d_ws; (void)ws_size; (void)out_size;
    const float* xyz     = (const float*)d_in[0];
    const float* new_xyz = (const float*)d_in[1];
    const float* feat    = (const float*)d_in[2];
    float* out = (float*)d_out;

    const int totalWaves = B_ * (N_ / 32);     // 1024 waves, 32 rows each
    dim3 block(128);                           // 4 waves per block
    dim3 grid(totalWaves / 4);                 // 256 blocks
    knn_interp_wmma<<<grid, block, 0, stream>>>(xyz, new_xyz, feat, out);
}